// GConv_57475252355660
// MI455X (gfx1250) — compile-verified
//
#include <hip/hip_runtime.h>
#include <hip/hip_bf16.h>
#include <stdint.h>

typedef float v2f __attribute__((ext_vector_type(2)));
typedef float v8f __attribute__((ext_vector_type(8)));

#define N_NODES 50000
#define N_EDGES 1600000
#define N_GRAPHS 500
#define DIM 128
#define LCAT 384            // L * DIM
#define BN_EPS 1e-5f

// ---------------------------------------------------------------------------
// zero a float buffer
__global__ void zero_f32(float* __restrict__ p, int n) {
    int i = blockIdx.x * blockDim.x + threadIdx.x;
    if (i < n) p[i] = 0.0f;
}

// ---------------------------------------------------------------------------
// agg[i, :] = z[i, :]   (z may be strided: layer 0 reads x (ld=128),
// later layers read the previous slice of z_cat (ld=384))
__global__ void init_agg(float* __restrict__ agg,
                         const float* __restrict__ z, int zld) {
    int i = blockIdx.x * blockDim.x + threadIdx.x;   // N_NODES * 32 threads
    int row = i >> 5;
    int c4  = (i & 31) << 2;
    float4 v = *(const float4*)(z + (size_t)row * zld + c4);
    *(float4*)(agg + (size_t)row * DIM + c4) = v;
}

// ---------------------------------------------------------------------------
// agg[dst, :] += z[src, :]   — 32 threads per edge, float4 gather + 4 atomics
__global__ void scatter_edges(float* __restrict__ agg,
                              const float* __restrict__ z, int zld,
                              const int* __restrict__ src,
                              const int* __restrict__ dst) {
    unsigned long long i = (unsigned long long)blockIdx.x * blockDim.x + threadIdx.x;
    int e  = (int)(i >> 5);
    int c4 = ((int)i & 31) << 2;
    int s = src[e];
    int d = dst[e];
    float4 v = *(const float4*)(z + (size_t)s * zld + c4);
    float* a = agg + (size_t)d * DIM + c4;
    atomicAdd(a + 0, v.x);
    atomicAdd(a + 1, v.y);
    atomicAdd(a + 2, v.z);
    atomicAdd(a + 3, v.w);
}

// ---------------------------------------------------------------------------
// Fused MLP: hpre = relu(agg @ W1 + b1) @ W2 + b2
// 256 threads = 8 wave32. HAS2=true: block handles 32 rows (two 16-row
// M-tiles sharing every B-operand load). HAS2=false: tail block, 16 rows.
// Wave w owns output columns [16w, 16w+16).
// GEMMs via V_WMMA_F32_16X16X4_F32, K swept in steps of 4; A tiles staged
// global->LDS with CDNA5 async loads (ASYNCcnt). Branch-free inner loops.
template <bool HAS2>
__global__ __launch_bounds__(256) void mlp_wmma(
    const float* __restrict__ agg, float* __restrict__ hpre,
    const float* __restrict__ W1, const float* __restrict__ b1,
    const float* __restrict__ W2, const float* __restrict__ b2,
    int row_base) {
    __shared__ float sA[32 * 132];   // padded stride 132: avoids 64-bank conflicts
    __shared__ float sH[32 * 132];

    const int tid  = threadIdx.x;
    const int wave = tid >> 5;
    const int lane = tid & 31;
    const int row0 = row_base + blockIdx.x * (HAS2 ? 32 : 16);
    constexpr int NVEC = HAS2 ? 1024 : 512;       // float4s to stage

    // ---- async global->LDS stage of the (32|16)x128 A tile ----------------
    {
        const float* gbase = agg + (size_t)row0 * DIM;
        const unsigned ldsA = (unsigned)(uintptr_t)sA;   // low 32 bits = LDS offset
        #pragma unroll
        for (int j = tid; j < NVEC; j += 256) {
            int m  = j >> 5;                     // row within tile
            int k4 = (j & 31) << 2;              // first of 4 floats
            unsigned loff = ldsA + (unsigned)(m * 132 + k4) * 4u;
            unsigned goff = (unsigned)j * 16u;   // byte offset into agg block
            asm volatile("global_load_async_to_lds_b128 %0, %1, %2"
                         :: "v"(loff), "v"(goff), "s"(gbase)
                         : "memory");
        }
        asm volatile("s_wait_asynccnt 0x0" ::: "memory");
    }
    __syncthreads();

    // A layout: lanes 0-15 and 16-31 both hold M=0..15; K halves differ by 2
    const int m     = lane & 15;
    const int khalf = (lane >> 4) << 1;           // 0 or 2
    const int col   = (wave << 4) + (lane & 15);  // output column of this lane
    const int mbase = (lane >> 4) << 3;           // D rows: 0..7 or 8..15

    // ---- GEMM1: sA @ W1, B register pair reused across both M-tiles -------
    v8f c1a = {}, c1b = {};
    #pragma unroll 4
    for (int k0 = 0; k0 < DIM; k0 += 4) {
        const int kb = k0 + khalf;
        v2f b; b.x = W1[(size_t)kb * DIM + col];
               b.y = W1[(size_t)(kb + 1) * DIM + col];
        v2f a0; a0.x = sA[m * 132 + kb];
                a0.y = sA[m * 132 + kb + 1];
        c1a = __builtin_amdgcn_wmma_f32_16x16x4_f32(
                  false, a0, false, b, (short)0, c1a, false, false);
        if constexpr (HAS2) {
            v2f a1; a1.x = sA[(m + 16) * 132 + kb];
                    a1.y = sA[(m + 16) * 132 + kb + 1];
            c1b = __builtin_amdgcn_wmma_f32_16x16x4_f32(
                      false, a1, false, b, (short)0, c1b, false, false);
        }
    }

    // bias + ReLU, stash H1 tiles to LDS for GEMM2's A operand
    {
        const float bias1 = b1[col];
        #pragma unroll
        for (int r = 0; r < 8; ++r) {
            float v = c1a[r] + bias1;
            sH[(mbase + r) * 132 + col] = v > 0.0f ? v : 0.0f;
        }
        if constexpr (HAS2) {
            #pragma unroll
            for (int r = 0; r < 8; ++r) {
                float v = c1b[r] + bias1;
                sH[(16 + mbase + r) * 132 + col] = v > 0.0f ? v : 0.0f;
            }
        }
    }
    __syncthreads();

    // ---- GEMM2: sH @ W2 ---------------------------------------------------
    v8f c2a = {}, c2b = {};
    #pragma unroll 4
    for (int k0 = 0; k0 < DIM; k0 += 4) {
        const int kb = k0 + khalf;
        v2f b; b.x = W2[(size_t)kb * DIM + col];
               b.y = W2[(size_t)(kb + 1) * DIM + col];
        v2f a0; a0.x = sH[m * 132 + kb];
                a0.y = sH[m * 132 + kb + 1];
        c2a = __builtin_amdgcn_wmma_f32_16x16x4_f32(
                  false, a0, false, b, (short)0, c2a, false, false);
        if constexpr (HAS2) {
            v2f a1; a1.x = sH[(m + 16) * 132 + kb];
                    a1.y = sH[(m + 16) * 132 + kb + 1];
            c2b = __builtin_amdgcn_wmma_f32_16x16x4_f32(
                      false, a1, false, b, (short)0, c2b, false, false);
        }
    }

    {
        const float bias2 = b2[col];
        #pragma unroll
        for (int r = 0; r < 8; ++r) {
            hpre[(size_t)(row0 + mbase + r) * DIM + col] = c2a[r] + bias2;
        }
        if constexpr (HAS2) {
            #pragma unroll
            for (int r = 0; r < 8; ++r) {
                hpre[(size_t)(row0 + 16 + mbase + r) * DIM + col] = c2b[r] + bias2;
            }
        }
    }
}

// ---------------------------------------------------------------------------
// column-wise sum / sum-of-squares over hpre [N, 128]
__global__ void bn_stats(const float* __restrict__ hpre,
                         float* __restrict__ sums,
                         float* __restrict__ sumsq) {
    const int col = threadIdx.x & 127;
    const int sub = threadIdx.x >> 7;         // 0 or 1
    const int r0  = blockIdx.x * 200;         // 250 blocks * 200 rows = 50000
    float s = 0.0f, s2 = 0.0f;
    for (int r = r0 + sub; r < r0 + 200; r += 2) {
        float v = hpre[(size_t)r * DIM + col];
        s  += v;
        s2 += v * v;
    }
    atomicAdd(&sums[col],  s);
    atomicAdd(&sumsq[col], s2);
}

// ---------------------------------------------------------------------------
// BN normalize (+ReLU except last layer), write z_cat slice, pool into g_cat
__global__ void bn_apply_pool(const float* __restrict__ hpre,
                              const float* __restrict__ sums,
                              const float* __restrict__ sumsq,
                              const float* __restrict__ gamma,
                              const float* __restrict__ beta,
                              const int* __restrict__ batch,
                              float* __restrict__ zcat,
                              float* __restrict__ gcat,
                              int lcol, int do_relu) {
    const int row = blockIdx.x;
    const int col = threadIdx.x;              // 128 threads
    const float inv_n = 1.0f / (float)N_NODES;
    float mu  = sums[col] * inv_n;
    float var = sumsq[col] * inv_n - mu * mu;
    float rs  = rsqrtf(var + BN_EPS);
    float v = (hpre[(size_t)row * DIM + col] - mu) * rs * gamma[col] + beta[col];
    if (do_relu) v = fmaxf(v, 0.0f);
    zcat[(size_t)row * LCAT + lcol + col] = v;
    atomicAdd(&gcat[(size_t)batch[row] * LCAT + lcol + col], v);
}

// ---------------------------------------------------------------------------
extern "C" void kernel_launch(void* const* d_in, const int* in_sizes, int n_in,
                              void* d_out, int out_size, void* d_ws, size_t ws_size,
                              hipStream_t stream) {
    const float* x     = (const float*)d_in[0];
    const int*   eidx  = (const int*)d_in[1];       // [2, E]
    const int*   batch = (const int*)d_in[2];
    const float* W1    = (const float*)d_in[3];     // [L, D, D]
    const float* b1    = (const float*)d_in[4];     // [L, D]
    const float* W2    = (const float*)d_in[5];
    const float* b2    = (const float*)d_in[6];
    const float* gamma = (const float*)d_in[7];
    const float* beta  = (const float*)d_in[8];

    const int* src = eidx;
    const int* dst = eidx + N_EDGES;

    float* z_cat = (float*)d_out;                                  // [N, 384]
    float* g_cat = z_cat + (size_t)N_NODES * LCAT;                 // [G, 384]

    float* agg   = (float*)d_ws;                                   // N*D
    float* hpre  = agg  + (size_t)N_NODES * DIM;                   // N*D
    float* sums  = hpre + (size_t)N_NODES * DIM;                   // D
    float* sumsq = sums + DIM;                                     // D

    // pooled output accumulated with atomics -> zero it every call
    zero_f32<<<(N_GRAPHS * LCAT + 255) / 256, 256, 0, stream>>>(g_cat, N_GRAPHS * LCAT);

    const int full_blocks = N_NODES / 32;          // 1562 blocks of 32 rows
    const int tail_row    = full_blocks * 32;      // 49984: one 16-row block

    for (int l = 0; l < 3; ++l) {
        const float* zptr = (l == 0) ? x : (z_cat + (size_t)(l - 1) * DIM);
        const int    zld  = (l == 0) ? DIM : LCAT;

        init_agg<<<(N_NODES * 32) / 256, 256, 0, stream>>>(agg, zptr, zld);
        scatter_edges<<<(N_EDGES * 32) / 256, 256, 0, stream>>>(agg, zptr, zld, src, dst);

        const float* W1l = W1 + (size_t)l * DIM * DIM;
        const float* b1l = b1 + (size_t)l * DIM;
        const float* W2l = W2 + (size_t)l * DIM * DIM;
        const float* b2l = b2 + (size_t)l * DIM;

        mlp_wmma<true><<<full_blocks, 256, 0, stream>>>(
            agg, hpre, W1l, b1l, W2l, b2l, 0);
        mlp_wmma<false><<<1, 256, 0, stream>>>(
            agg, hpre, W1l, b1l, W2l, b2l, tail_row);

        zero_f32<<<1, 256, 0, stream>>>(sums, 2 * DIM);  // sums+sumsq contiguous
        bn_stats<<<250, 256, 0, stream>>>(hpre, sums, sumsq);

        bn_apply_pool<<<N_NODES, DIM, 0, stream>>>(
            hpre, sums, sumsq,
            gamma + (size_t)l * DIM, beta + (size_t)l * DIM,
            batch, z_cat, g_cat, l * DIM, (l != 2) ? 1 : 0);
    }
}